// PointPillarScatter3d_4080218931379
// MI455X (gfx1250) — compile-verified
//
#include <hip/hip_runtime.h>
#include <hip/hip_bf16.h>

// PointPillarScatter3d for MI455X (gfx1250).
//
// Pure scatter, 0 FLOPs. ~191 MB of mandatory HBM traffic => ~8.2 us at
// 23.3 TB/s. Strategy exploits the 192 MB L2: the whole 112 MB output is
// L2-resident, so random dword scatter-stores merge into full lines in L2
// and are written back to HBM exactly once. Streaming inputs use
// non-temporal hints so they do not evict the resident output.
//
// The per-pillar 512 B feature-row fetch is routed through the CDNA5 async
// copy path (global_load_async_to_lds_b128 + s_wait_asynccnt + ds_load_b128):
// identical global traffic, LDS round-trip is noise vs per-WGP LDS bandwidth,
// and latency is hidden by 8 waves/block.

#define NXc 468
#define NYc 468
#define NZc 1
#define CBc 128                      // features per pillar
#define NCELLS (NZc * NYc * NXc)     // 219024 cells

typedef __attribute__((ext_vector_type(4))) float v4f;
typedef __attribute__((ext_vector_type(4))) int   v4i;

// ---------------------------------------------------------------------------
// Kernel 1: zero-fill the dense grid with 128-bit stores (global_store_b128).
// Regular temporal: these lines are re-touched by the scatter while still
// resident in the 192 MB L2, so each line costs one HBM writeback total.
// ---------------------------------------------------------------------------
__global__ void pps_zero_fill(float* __restrict__ out, int n, int n4) {
    const int i = blockIdx.x * blockDim.x + threadIdx.x;
    if (i < n4) {
        v4f z = {0.0f, 0.0f, 0.0f, 0.0f};
        *((v4f*)out + i) = z;
    }
    if (i == 0) {                        // generic scalar tail
        for (int k = n4 * 4; k < n; ++k) out[k] = 0.0f;
    }
}

// ---------------------------------------------------------------------------
// Kernel 2: scatter. One wave32 per pillar; lane L owns features 4L..4L+3.
//  - coords: NT b128 load (wave-uniform address, cache-broadcast)
//  - features: per-lane async 16B global->LDS copy (one wave = 512 B
//    contiguous), drained with s_wait_asynccnt, read back via ds_load_b128
//  - output: 4 scattered dword stores per lane, merged in L2
//  - global_prefetch_b8 for rows 8 pillars ahead
// All early exits are wave-uniform, so EXEC is full at the async op.
// ---------------------------------------------------------------------------
__global__ void pps_scatter(const float* __restrict__ feats,
                            const int*   __restrict__ coords,
                            float*       __restrict__ out,
                            int n_pillars) {
    // 8 waves per 256-thread block, 512 B LDS slot per wave.
    __shared__ __align__(16) float smem[8 * CBc];

    const int tid  = blockIdx.x * blockDim.x + threadIdx.x;
    const int p    = tid >> 5;               // pillar index (one wave per pillar)
    const int lane = tid & 31;
    const int wave = (threadIdx.x >> 5) & 7;
    if (p >= n_pillars) return;              // wave-uniform

    // Prefetch the feature row this lane will need 8 pillars ahead
    // (lowers to global_prefetch_b8 on gfx1250).
    if (p + 8 < n_pillars) {
        __builtin_prefetch(feats + (size_t)(p + 8) * CBc + lane * 4, 0, 1);
    }

    // coords[p] = {b, z, y, x}; non-temporal streaming load.
    const v4i c4 = __builtin_nontemporal_load((const v4i*)coords + p);
    if (c4.x != 0) return;                   // batch mask (wave-uniform)

    const int idx = c4.y * (NYc * NXc) + c4.z * NXc + c4.w;
    if ((unsigned)idx >= (unsigned)NCELLS) return;  // drop semantics (uniform)

    // ---- async copy: 16 B per lane, global -> LDS (ASYNCcnt-tracked) ----
    float* slot = &smem[wave * CBc + lane * 4];
    const unsigned lds_off =
        (unsigned)(size_t)(__attribute__((address_space(3))) float*)slot;
    const unsigned long long gaddr =
        (unsigned long long)(feats + (size_t)p * CBc + lane * 4);

    asm volatile("global_load_async_to_lds_b128 %0, %1, off"
                 :: "v"(lds_off), "v"(gaddr)
                 : "memory");
    asm volatile("s_wait_asynccnt 0" ::: "memory");

    // Read back this lane's 16 B slot (ds_load_b128).
    const v4f f = *(const v4f*)slot;

    const int c = lane * 4;
    float* base = out + idx;
    // Scattered dword stores, plane stride NCELLS. Regular temporal so they
    // merge with the zero-filled lines in L2 (one HBM writeback per line).
    base[(size_t)(c + 0) * NCELLS] = f.x;
    base[(size_t)(c + 1) * NCELLS] = f.y;
    base[(size_t)(c + 2) * NCELLS] = f.z;
    base[(size_t)(c + 3) * NCELLS] = f.w;
}

extern "C" void kernel_launch(void* const* d_in, const int* in_sizes, int n_in,
                              void* d_out, int out_size, void* d_ws, size_t ws_size,
                              hipStream_t stream) {
    const float* feats  = (const float*)d_in[0];   // [N_PILLARS, 128] f32
    const int*   coords = (const int*)d_in[1];     // [N_PILLARS, 4]   i32
    float*       out    = (float*)d_out;           // [1, 128, 468, 468] f32

    const int n_pillars = in_sizes[0] / CBc;       // 150000

    // --- zero fill ---
    const int n4    = out_size >> 2;               // float4 count
    const int zblk  = 256;
    const int zgrid = (n4 + zblk - 1) / zblk;
    pps_zero_fill<<<zgrid, zblk, 0, stream>>>(out, out_size, n4);

    // --- scatter: 32 threads (one wave) per pillar, 8 waves per block ---
    const long long threads = (long long)n_pillars * 32;
    const int sblk  = 256;
    const int sgrid = (int)((threads + sblk - 1) / sblk);
    pps_scatter<<<sgrid, sblk, 0, stream>>>(feats, coords, out, n_pillars);
}